// GTO_Expansion_9216999817548
// MI455X (gfx1250) — compile-verified
//
#include <hip/hip_runtime.h>
#include <math.h>

typedef __attribute__((ext_vector_type(2))) float v2f;
typedef __attribute__((ext_vector_type(8))) float v8f;

#define NNODES 50000
#define NEDGES 800000
#define NB 32
#define NA 128
#define TT 6

// silu via v_exp_f32 + v_rcp_f32 (1-ulp approx reciprocal) instead of the
// full IEEE divide expansion (v_div_scale + Newton chain, ~10 VALU ops).
__device__ __forceinline__ float silu_f(float x) {
  return x * __builtin_amdgcn_rcpf(1.0f + __expf(-x));
}

__device__ __forceinline__ v8f wmma4(v2f a, v2f b, v8f c) {
  // V_WMMA_F32_16X16X4_F32: D = A(16x4) * B(4x16) + C(16x16), full f32
  return __builtin_amdgcn_wmma_f32_16x16x4_f32(false, a, false, b, (short)0, c,
                                               false, false);
}

// ---------------------------------------------------------------------------
// cj = silu(feat @ cj_w1 + cj_b1) @ cj_w2 + cj_b2      [N,1]
// One wave per 16-node tile. 8 column tiles x 32 K-steps of WMMA f32 16x16x4.
// ---------------------------------------------------------------------------
__global__ __launch_bounds__(32) void cj_kernel(
    const float* __restrict__ feat, const float* __restrict__ w1,
    const float* __restrict__ b1, const float* __restrict__ w2,
    const float* __restrict__ b2, float* __restrict__ cj) {
  const int lane = threadIdx.x;
  const int half = lane >> 4;
  const int ml = lane & 15;
  const int base = blockIdx.x * 16;
  const float* arow = feat + (size_t)(base + ml) * NA;

  float acc[8];
#pragma unroll
  for (int r = 0; r < 8; ++r) acc[r] = 0.f;

  for (int nt = 0; nt < 8; ++nt) {
    const int col = nt * 16 + ml;
    v8f c = {0.f, 0.f, 0.f, 0.f, 0.f, 0.f, 0.f, 0.f};
#pragma unroll 4
    for (int k = 0; k < NA; k += 4) {
      v2f a, b;
      a.x = arow[k + 2 * half];
      a.y = arow[k + 2 * half + 1];
      b.x = w1[(size_t)(k + 2 * half) * NA + col];
      b.y = w1[(size_t)(k + 2 * half + 1) * NA + col];
      c = wmma4(a, b, c);
    }
    const float b1c = b1[col];
    const float w2c = w2[col];
#pragma unroll
    for (int r = 0; r < 8; ++r) acc[r] += silu_f(c[r] + b1c) * w2c;
  }
  // reduce over the 16 N-columns held by each half-wave
#pragma unroll
  for (int m = 1; m <= 8; m <<= 1)
#pragma unroll
    for (int r = 0; r < 8; ++r) acc[r] += __shfl_xor(acc[r], m, 32);
  if (ml == 0) {
    const float bias = b2[0];
#pragma unroll
    for (int r = 0; r < 8; ++r) cj[base + half * 8 + r] = acc[r] + bias;
  }
}

// ---------------------------------------------------------------------------
__global__ void zero_kernel(float* __restrict__ p, int n) {
  int i = blockIdx.x * blockDim.x + threadIdx.x;
  if (i < n) p[i] = 0.f;
}

// ---------------------------------------------------------------------------
// Per (edge, basis): gauss[6], scatter fij into agg via HW f32 atomics,
// and the tiny 6x6 filter MLP -> filt[E,32]. One wave covers all 32 basis
// slots of one edge, so the 6*32=192 atomic targets per wave are contiguous.
// ---------------------------------------------------------------------------
__global__ __launch_bounds__(256) void edge_kernel(
    const float* __restrict__ dis_vec, const float* __restrict__ cj,
    const float* __restrict__ fw1, const float* __restrict__ fw2,
    const float* __restrict__ fb2, const int* __restrict__ src,
    const int* __restrict__ dst, float* __restrict__ agg,
    float* __restrict__ filt) {
  const long long idx = (long long)blockIdx.x * blockDim.x + threadIdx.x;
  if (idx >= (long long)NEDGES * NB) return;
  const int e = (int)(idx >> 5);
  const int b = (int)(idx & 31);

  const float vx = dis_vec[3 * (size_t)e + 0];
  const float vy = dis_vec[3 * (size_t)e + 1];
  const float vz = dis_vec[3 * (size_t)e + 2];

  // signed_pow with integer exponents 0/1/2 reduces to {1, x, x*x}
  const float xs = vx + 1e-8f, ys = vy + 1e-8f, zs = vz + 1e-8f;
  const float px[3] = {1.f, xs, xs * xs};
  const float py[3] = {1.f, ys, ys * ys};
  const float pz[3] = {1.f, zs, zs * zs};

  const int IP[TT] = {0, 0, 0, 1, 1, 2};
  const int JP[TT] = {0, 1, 2, 0, 1, 0};
  const int KP[TT] = {2, 1, 0, 1, 0, 0};
  const float P2[TT] = {1.f, 1.41421356f, 1.f, 1.41421356f, 1.41421356f, 1.f};

  const float dx = vx + 1e-9f, dy = vy + 1e-9f, dz = vz + 1e-9f;
  const float dis = sqrtf(dx * dx + dy * dy + dz * dz);
  const float width = 10.0f / 31.0f;
  const float coeff = -0.5f / (width * width);
  const float off = 10.0f * (float)b / 31.0f;
  const float dd = dis - off;
  const float g0 = __expf(coeff * dd * dd);

  float g[TT];
#pragma unroll
  for (int t = 0; t < TT; ++t) g[t] = px[IP[t]] * py[JP[t]] * pz[KP[t]] * P2[t] * g0;

  // scatter fij = gauss * cj[dst]  into  agg[src][b][t]  (L2-resident atomics)
  const float cjd = cj[dst[e]];
  float* ap = agg + (size_t)src[e] * (NB * TT) + b * TT;
#pragma unroll
  for (int t = 0; t < TT; ++t) unsafeAtomicAdd(ap + t, g[t] * cjd);

  // gn = gauss / (||gauss + 1e-8||_T + 1)
  float s = 0.f;
#pragma unroll
  for (int t = 0; t < TT; ++t) {
    const float u = g[t] + 1e-8f;
    s += u * u;
  }
  const float inv = __builtin_amdgcn_rcpf(sqrtf(s) + 1.f);

  // filt = silu(gn @ fw1) @ fw2 + fb2
  float fv = fb2[0];
#pragma unroll
  for (int j = 0; j < TT; ++j) {
    float h = 0.f;
#pragma unroll
    for (int t = 0; t < TT; ++t) h += g[t] * fw1[t * TT + j];
    fv += silu_f(h * inv) * fw2[j];
  }
  filt[(size_t)e * NB + b] = fv;
}

// ---------------------------------------------------------------------------
// msg_out = silu(msg_in @ w1 + b1) @ w2 + b2, msg_in from agg.
// One wave per 16-node tile; GEMM1 -> silu -> GEMM2 chained through LDS.
// ---------------------------------------------------------------------------
__global__ __launch_bounds__(32) void msg_kernel(
    const float* __restrict__ agg, const float* __restrict__ w1,
    const float* __restrict__ b1, const float* __restrict__ w2,
    const float* __restrict__ b2, float* __restrict__ out) {
  __shared__ float s_in[16][NB];    // msg_in tile (A of GEMM1)
  __shared__ float s_hid[16][NA];   // hidden tile (A of GEMM2)
  const int lane = threadIdx.x;
  const int half = lane >> 4;
  const int ml = lane & 15;
  const int base = blockIdx.x * 16;

  // f2[n][b] = sum_t agg[n][b][t]^2 ; msg_in = f2 / (||f2+1e-9||_B + 1)
  for (int i = 0; i < 16; ++i) {
    const float* ap = agg + ((size_t)(base + i) * NB + lane) * TT;
    float f2 = 0.f;
#pragma unroll
    for (int t = 0; t < TT; ++t) {
      const float v = ap[t];
      f2 += v * v;
    }
    float u = f2 + 1e-9f;
    float sq = u * u;
#pragma unroll
    for (int m = 1; m < 32; m <<= 1) sq += __shfl_xor(sq, m, 32);
    s_in[i][lane] = f2 * __builtin_amdgcn_rcpf(sqrtf(sq) + 1.f);
  }
  __syncthreads();

  // GEMM1: hidden = silu(msg_in[16,32] @ w1[32,128] + b1)
  for (int nt = 0; nt < 8; ++nt) {
    const int col = nt * 16 + ml;
    v8f c = {0.f, 0.f, 0.f, 0.f, 0.f, 0.f, 0.f, 0.f};
#pragma unroll
    for (int k = 0; k < NB; k += 4) {
      v2f a, bb;
      a.x = s_in[ml][k + 2 * half];
      a.y = s_in[ml][k + 2 * half + 1];
      bb.x = w1[(size_t)(k + 2 * half) * NA + col];
      bb.y = w1[(size_t)(k + 2 * half + 1) * NA + col];
      c = wmma4(a, bb, c);
    }
    const float b1c = b1[col];
#pragma unroll
    for (int r = 0; r < 8; ++r) s_hid[r + 8 * half][col] = silu_f(c[r] + b1c);
  }
  __syncthreads();

  // GEMM2: out = hidden[16,128] @ w2[128,128] + b2
  for (int nt = 0; nt < 8; ++nt) {
    const int col = nt * 16 + ml;
    v8f c = {0.f, 0.f, 0.f, 0.f, 0.f, 0.f, 0.f, 0.f};
#pragma unroll 4
    for (int k = 0; k < NA; k += 4) {
      v2f a, bb;
      a.x = s_hid[ml][k + 2 * half];
      a.y = s_hid[ml][k + 2 * half + 1];
      bb.x = w2[(size_t)(k + 2 * half) * NA + col];
      bb.y = w2[(size_t)(k + 2 * half + 1) * NA + col];
      c = wmma4(a, bb, c);
    }
    const float b2c = b2[col];
#pragma unroll
    for (int r = 0; r < 8; ++r)
      out[(size_t)(base + r + 8 * half) * NA + col] = c[r] + b2c;
  }
}

// ---------------------------------------------------------------------------
extern "C" void kernel_launch(void* const* d_in, const int* in_sizes, int n_in,
                              void* d_out, int out_size, void* d_ws,
                              size_t ws_size, hipStream_t stream) {
  const float* feat = (const float*)d_in[0];
  const float* dis_vec = (const float*)d_in[1];
  const float* cj_w1 = (const float*)d_in[2];
  const float* cj_b1 = (const float*)d_in[3];
  const float* cj_w2 = (const float*)d_in[4];
  const float* cj_b2 = (const float*)d_in[5];
  const float* msg_w1 = (const float*)d_in[6];
  const float* msg_b1 = (const float*)d_in[7];
  const float* msg_w2 = (const float*)d_in[8];
  const float* msg_b2 = (const float*)d_in[9];
  const float* filt_w1 = (const float*)d_in[10];
  const float* filt_w2 = (const float*)d_in[11];
  const float* filt_b2 = (const float*)d_in[12];
  const int* src = (const int*)d_in[13];
  const int* dst = (const int*)d_in[14];

  float* msg_out = (float*)d_out;                       // [N,128]
  float* filt_out = msg_out + (size_t)NNODES * NA;      // [E,32]

  float* cj = (float*)d_ws;                             // [N]
  float* agg = cj + 50176;                              // [N,32,6], 38.4 MB

  const int aggN = NNODES * NB * TT;  // 9,600,000
  zero_kernel<<<(aggN + 255) / 256, 256, 0, stream>>>(agg, aggN);

  cj_kernel<<<NNODES / 16, 32, 0, stream>>>(feat, cj_w1, cj_b1, cj_w2, cj_b2,
                                            cj);

  const long long ethreads = (long long)NEDGES * NB;  // 25.6M
  edge_kernel<<<(int)((ethreads + 255) / 256), 256, 0, stream>>>(
      dis_vec, cj, filt_w1, filt_w2, filt_b2, src, dst, agg, filt_out);

  msg_kernel<<<NNODES / 16, 32, 0, stream>>>(agg, msg_w1, msg_b1, msg_w2,
                                             msg_b2, msg_out);
}